// ProposalLayer_37761352466516
// MI455X (gfx1250) — compile-verified
//
#include <hip/hip_runtime.h>
#include <hip/hip_bf16.h>
#include <math.h>

// ---------------- problem constants (match reference) ----------------
#define B_     8
#define N_     262144
#define KPRE   6000        // PRE_NMS_LIMIT
#define KOUT   1000        // PROPOSAL_COUNTS
#define IOUTHR 0.7f
#define BINS   4096        // score histogram bins over (0,1) (softmax output)
#define CAP    8192        // candidate cap (power of two for bitonic sort)

// ---------------- workspace layout (bytes) ----------------
static constexpr size_t OFF_HIST = 0;                              // B*BINS u32
static constexpr size_t OFF_CNT  = OFF_HIST + (size_t)B_*BINS*4;   // B u32
static constexpr size_t OFF_THR  = OFF_CNT  + 64;                  // B u32
static constexpr size_t OFF_NV   = OFF_THR  + 64;                  // B u32
static constexpr size_t OFF_KEYS = OFF_NV   + 64;                  // B*CAP f32
static constexpr size_t OFF_IDXS = OFF_KEYS + (size_t)B_*CAP*4;    // B*CAP i32
static constexpr size_t OFF_BOX  = OFF_IDXS + (size_t)B_*CAP*4;    // B*KPRE*4 f32

// ---------------- CDNA5 async global->LDS copy (gfx1250) ----------------
// LDS address comes from the vdst VGPR; global addr = SGPR64 base + VGPR32 offset (GVS).
__device__ __forceinline__ void async_copy_b128(unsigned lds_off, unsigned goff,
                                                const void* base) {
  asm volatile("global_load_async_to_lds_b128 %0, %1, %2"
               :: "v"(lds_off), "v"(goff), "s"(base)
               : "memory");
}
__device__ __forceinline__ void wait_async0() {
  asm volatile("s_wait_asynccnt 0" ::: "memory");
}

// ---------------- k0: zero histograms + counters ----------------
__global__ void k_init(unsigned* ws_u32, int nwords) {
  for (int i = blockIdx.x * blockDim.x + threadIdx.x; i < nwords;
       i += gridDim.x * blockDim.x)
    ws_u32[i] = 0u;
}

// ---------------- k1: per-batch score histogram (LDS atomics) ----------------
__global__ void k_hist(const float2* __restrict__ rc, unsigned* __restrict__ hist) {
  __shared__ unsigned lh[BINS];
  const int b = blockIdx.x >> 4;          // 16 blocks per batch
  const int chunk = blockIdx.x & 15;
  const int CHUNK = N_ / 16;
  for (int i = threadIdx.x; i < BINS; i += blockDim.x) lh[i] = 0u;
  __syncthreads();
  const size_t base = (size_t)b * N_ + (size_t)chunk * CHUNK;
  for (int i = threadIdx.x; i < CHUNK; i += blockDim.x) {
    __builtin_prefetch(&rc[base + i + 4096], 0, 1);   // global_prefetch_b8
    float s = rc[base + i].y;                         // rpn_class[...,1]
    int bin = (int)(s * (float)BINS);
    bin = bin < 0 ? 0 : (bin > BINS - 1 ? BINS - 1 : bin);
    atomicAdd(&lh[bin], 1u);
  }
  __syncthreads();
  for (int i = threadIdx.x; i < BINS; i += blockDim.x) {
    unsigned c = lh[i];
    if (c) atomicAdd(&hist[(size_t)b * BINS + i], c);
  }
}

// ---------------- k2: find per-batch threshold bin ----------------
__global__ void k_thresh(const unsigned* __restrict__ hist, unsigned* __restrict__ thr) {
  const int b = blockIdx.x;
  if (threadIdx.x != 0) return;
  unsigned cum = 0; int t = 0;
  for (int bin = BINS - 1; bin >= 0; --bin) {
    cum += hist[(size_t)b * BINS + bin];
    if (cum >= KPRE) { t = bin; break; }
  }
  // keep candidate count under CAP (sort buffer size)
  while (cum > CAP && t < BINS - 1) { cum -= hist[(size_t)b * BINS + t]; t++; }
  thr[b] = (unsigned)t;
}

// ---------------- k3: compact candidates (score, index) ----------------
__global__ void k_compact(const float2* __restrict__ rc, const unsigned* __restrict__ thr,
                          unsigned* __restrict__ cnt, float* __restrict__ keys,
                          int* __restrict__ idxs) {
  const long long total = (long long)B_ * N_;
  for (long long g = (long long)blockIdx.x * blockDim.x + threadIdx.x; g < total;
       g += (long long)gridDim.x * blockDim.x) {
    int b = (int)(g / N_);
    int i = (int)(g - (long long)b * N_);
    float s = rc[g].y;
    int bin = (int)(s * (float)BINS);
    bin = bin < 0 ? 0 : (bin > BINS - 1 ? BINS - 1 : bin);
    if ((unsigned)bin >= thr[b]) {
      unsigned pos = atomicAdd(&cnt[b], 1u);
      if (pos < CAP) {
        keys[(size_t)b * CAP + pos] = s;
        idxs[(size_t)b * CAP + pos] = i;
      }
    }
  }
}

// ---------------- k4: bitonic sort (desc score, asc idx) + decode boxes ----------
__global__ void k_sortdecode(const float* __restrict__ rpn_bbox,
                             const float* __restrict__ anchors,
                             const unsigned* __restrict__ cnt,
                             const float* __restrict__ keys,
                             const int* __restrict__ idxs,
                             float* __restrict__ boxes,   // B*KPRE*4 (clipped, sorted)
                             unsigned* __restrict__ nvalid) {
  extern __shared__ char smem[];
  float* sk = (float*)smem;            // CAP floats
  int*   si = (int*)(smem + CAP * 4);  // CAP ints
  const int b = blockIdx.x;
  const int n = (int)min(cnt[b], (unsigned)CAP);

  for (int t = threadIdx.x; t < CAP; t += blockDim.x) {
    if (t < n) { sk[t] = keys[(size_t)b * CAP + t]; si[t] = idxs[(size_t)b * CAP + t]; }
    else       { sk[t] = -3.0e38f;                  si[t] = 0x7fffffff; }
  }
  __syncthreads();

  for (unsigned k = 2; k <= CAP; k <<= 1) {
    for (unsigned j = k >> 1; j > 0; j >>= 1) {
      for (unsigned t = threadIdx.x; t < CAP; t += blockDim.x) {
        unsigned x = t ^ j;
        if (x > t) {
          float ka = sk[t], kb = sk[x];
          int   ia = si[t], ib = si[x];
          // c1: element at x should come BEFORE element at t (higher score, tie: lower idx)
          bool c1 = (kb > ka) || (kb == ka && ib < ia);
          bool up = ((t & k) == 0);
          if (c1 == up) { sk[t] = kb; sk[x] = ka; si[t] = ib; si[x] = ia; }
        }
      }
      __syncthreads();
    }
  }

  const int nv = n < KPRE ? n : KPRE;
  if (threadIdx.x == 0) nvalid[b] = (unsigned)nv;

  const float4* an4 = (const float4*)anchors;
  const float4* bb4 = (const float4*)rpn_bbox;
  float4* out4 = (float4*)boxes;
  for (int t = threadIdx.x; t < KPRE; t += blockDim.x) {
    float4 r = make_float4(0.f, 0.f, 0.f, 0.f);
    if (t < nv) {
      int idx = si[t];
      float4 a = an4[(size_t)b * N_ + idx];
      float4 d = bb4[(size_t)b * N_ + idx];
      d.x *= 0.1f; d.y *= 0.1f; d.z *= 0.2f; d.w *= 0.2f;  // RPN_BBOX_STD_DEV
      float hh = a.z - a.x, ww = a.w - a.y;
      float cy = a.x + 0.5f * hh, cx = a.y + 0.5f * ww;
      cy += d.x * hh; cx += d.y * ww;
      hh *= __expf(d.z); ww *= __expf(d.w);
      r.x = fminf(fmaxf(cy - 0.5f * hh, 0.f), 1.f);
      r.y = fminf(fmaxf(cx - 0.5f * ww, 0.f), 1.f);
      r.z = fminf(fmaxf(cy + 0.5f * hh, 0.f), 1.f);
      r.w = fminf(fmaxf(cx + 0.5f * ww, 0.f), 1.f);
    }
    out4[(size_t)b * KPRE + t] = r;
  }
}

// ---------------- k5: greedy NMS, boxes resident in LDS ----------------
__global__ void k_nms(const float* __restrict__ boxes, const unsigned* __restrict__ nvalid,
                      float* __restrict__ out) {
  extern __shared__ char smem[];
  float4*  sb  = (float4*)smem;                    // KPRE boxes  (96000 B)
  unsigned* sup = (unsigned*)(smem + (size_t)KPRE * 16); // KPRE flags (24000 B)
  __shared__ int s_next, s_pick;
  __shared__ float4 s_box;

  const int b  = blockIdx.x;
  const int nv = (int)nvalid[b];
  const float* base = boxes + (size_t)b * KPRE * 4;
  const unsigned sb_off = (unsigned)(size_t)(void*)sb;   // LDS byte offset

  // CDNA5 async tensor-path copy: global -> LDS, 16B per lane per issue
  for (int j = threadIdx.x; j < KPRE; j += blockDim.x) {
    async_copy_b128(sb_off + (unsigned)j * 16u, (unsigned)j * 16u, base);
    sup[j] = 0u;
  }
  wait_async0();
  __syncthreads();
  if (threadIdx.x == 0) s_next = 0;
  __syncthreads();

  float4* o4 = (float4*)(out + (size_t)b * KOUT * 4);
  for (int it = 0; it < KOUT; ++it) {
    if (threadIdx.x == 0) {
      int nx = s_next;
      while (nx < nv && sup[nx]) nx++;
      if (nx < nv) { s_pick = nx; sup[nx] = 1u; s_next = nx + 1; s_box = sb[nx]; }
      else         { s_pick = -1; s_next = nx; }
    }
    __syncthreads();
    const int pick = s_pick;
    if (pick >= 0) {
      const float4 pb = s_box;
      if (threadIdx.x == 0) o4[it] = pb;
      const float pa = (pb.z - pb.x) * (pb.w - pb.y);
      for (int j = threadIdx.x; j < nv; j += blockDim.x) {
        if (!sup[j]) {
          float4 q = sb[j];
          float y1 = fmaxf(pb.x, q.x), x1 = fmaxf(pb.y, q.y);
          float y2 = fminf(pb.z, q.z), x2 = fminf(pb.w, q.w);
          float inter = fmaxf(y2 - y1, 0.f) * fmaxf(x2 - x1, 0.f);
          float qa = (q.z - q.x) * (q.w - q.y);
          float iou = inter / (pa + qa - inter + 1e-9f);
          if (iou > IOUTHR) sup[j] = 1u;
        }
      }
    } else {
      if (threadIdx.x == 0) o4[it] = make_float4(0.f, 0.f, 0.f, 0.f);
    }
    __syncthreads();
  }
}

// ---------------- host-side launch ----------------
extern "C" void kernel_launch(void* const* d_in, const int* in_sizes, int n_in,
                              void* d_out, int out_size, void* d_ws, size_t ws_size,
                              hipStream_t stream) {
  (void)in_sizes; (void)n_in; (void)out_size; (void)ws_size;
  const float2* rc  = (const float2*)d_in[0];   // rpn_class (B,N,2) read as float2
  const float*  rb  = (const float*)d_in[1];    // rpn_bbox  (B,N,4)
  const float*  an  = (const float*)d_in[2];    // anchors   (B,N,4)
  float* out = (float*)d_out;                   // (B,1000,4)

  char* ws = (char*)d_ws;
  unsigned* hist = (unsigned*)(ws + OFF_HIST);
  unsigned* cnt  = (unsigned*)(ws + OFF_CNT);
  unsigned* thr  = (unsigned*)(ws + OFF_THR);
  unsigned* nval = (unsigned*)(ws + OFF_NV);
  float*    keys = (float*)(ws + OFF_KEYS);
  int*      idxs = (int*)(ws + OFF_IDXS);
  float*    bxs  = (float*)(ws + OFF_BOX);

  const int zwords = (int)(OFF_KEYS / 4);
  k_init<<<(zwords + 255) / 256, 256, 0, stream>>>((unsigned*)ws, zwords);
  k_hist<<<B_ * 16, 1024, 0, stream>>>(rc, hist);
  k_thresh<<<B_, 32, 0, stream>>>(hist, thr);
  k_compact<<<2048, 256, 0, stream>>>(rc, thr, cnt, keys, idxs);
  k_sortdecode<<<B_, 1024, (size_t)CAP * 8, stream>>>(rb, an, cnt, keys, idxs, bxs, nval);
  k_nms<<<B_, 1024, (size_t)KPRE * 20, stream>>>(bxs, nval, out);
}